// Attn_cid_time_90795608637908
// MI455X (gfx1250) — compile-verified
//
#include <hip/hip_runtime.h>
#include <math.h>

// Problem constants from the reference (fixed by setup_inputs()).
#define SEQ_LEN   4096      // history length == softmax width
#define N_TIME_C  10000     // cid_time row length
#define BLOCK     256       // 8 waves of 32
#define PER_TH    (SEQ_LEN / BLOCK)   // 16 values per thread, held in VGPRs
#define NWAVES    (BLOCK / 32)

typedef __attribute__((address_space(3))) char lds_char_t;

__device__ __forceinline__ float wave_reduce_max(float x) {
#pragma unroll
  for (int off = 16; off > 0; off >>= 1)
    x = fmaxf(x, __shfl_xor(x, off, 32));
  return x;
}

__device__ __forceinline__ float wave_reduce_sum(float x) {
#pragma unroll
  for (int off = 16; off > 0; off >>= 1)
    x += __shfl_xor(x, off, 32);
  return x;
}

__global__ __launch_bounds__(BLOCK)
void attn_gather_softmax_kernel(const int*   __restrict__ history,
                                const int*   __restrict__ current,
                                const float* __restrict__ cid_time,
                                float*       __restrict__ out,
                                int n_time)
{
  // Only the gathered row lives in LDS: ~40 KB -> 8 workgroups per 320 KB WGP.
  __shared__ __align__(16) float row_lds[N_TIME_C];
  __shared__ float red_max[NWAVES];
  __shared__ float red_sum[NWAVES];

  const int tid = threadIdx.x;
  const int row = blockIdx.x;

  // ---- load this thread's 16 history indices straight into registers ----
  // history is 16 KB shared by every block -> permanently L2-hot; these
  // coalesced loads issue before the async DMA and overlap with it.
  int hidx[PER_TH];
#pragma unroll
  for (int k = 0; k < PER_TH; ++k)
    hidx[k] = history[tid + k * BLOCK];

  // ---- async bulk copy of the selected cid_time row into LDS ----
  // 40000 B row, 16B-aligned (40000 % 16 == 0), 2500 b128 chunks.
  {
    const int cid = current[row];
    const unsigned long long gbase =
        (unsigned long long)(uintptr_t)(cid_time) +
        (unsigned long long)cid * (unsigned long long)n_time * 4ull;
    const unsigned lbase = (unsigned)(uintptr_t)(lds_char_t*)row_lds;

    for (int c = tid; c < N_TIME_C / 4; c += BLOCK) {
      const unsigned long long gaddr = gbase + (unsigned long long)c * 16ull;
      const unsigned           laddr = lbase + (unsigned)c * 16u;
      // CDNA5 async DMA: memory -> LDS, tracked by ASYNCcnt, no VGPR data.
      asm volatile("global_load_async_to_lds_b128 %0, %1, off"
                   :: "v"(laddr), "v"(gaddr)
                   : "memory");
    }
    // Drain this wave's outstanding async ops before signaling the barrier.
    asm volatile("s_wait_asynccnt 0" ::: "memory");
  }
  __syncthreads();

  // ---- pass 1: gather from LDS into registers, local max ----
  float v[PER_TH];
  float m = -INFINITY;
#pragma unroll
  for (int k = 0; k < PER_TH; ++k) {
    v[k] = row_lds[hidx[k]];                // single ds_load_b32 per element
    m = fmaxf(m, v[k]);
  }

  // block max reduction (wave32 shuffle + cross-wave LDS)
  m = wave_reduce_max(m);
  const int wid = tid >> 5;
  if ((tid & 31) == 0) red_max[wid] = m;
  __syncthreads();
  float bmax = red_max[0];
#pragma unroll
  for (int w = 1; w < NWAVES; ++w) bmax = fmaxf(bmax, red_max[w]);

  // ---- pass 2: exp in registers, local sum ----
  float s = 0.0f;
#pragma unroll
  for (int k = 0; k < PER_TH; ++k) {
    const float e = __expf(v[k] - bmax);    // v_exp_f32 path
    v[k] = e;
    s += e;
  }
  s = wave_reduce_sum(s);
  if ((tid & 31) == 0) red_sum[wid] = s;
  __syncthreads();
  float bsum = 0.0f;
#pragma unroll
  for (int w = 0; w < NWAVES; ++w) bsum += red_sum[w];

  const float inv = 1.0f / bsum;

  // ---- pass 3: normalized, coalesced, non-temporal stores ----
  // out is written strided (j = tid + k*256) -> fully coalesced b32 stores;
  // NT hint keeps the 64 MB streaming output from evicting cid_time in L2.
  float* orow = out + (size_t)row * SEQ_LEN;
#pragma unroll
  for (int k = 0; k < PER_TH; ++k) {
    const int j = tid + k * BLOCK;
    __builtin_nontemporal_store(v[k] * inv, &orow[j]);
  }
}

extern "C" void kernel_launch(void* const* d_in, const int* in_sizes, int n_in,
                              void* d_out, int out_size, void* d_ws, size_t ws_size,
                              hipStream_t stream) {
  (void)n_in; (void)d_ws; (void)ws_size; (void)out_size;
  const int*   history  = (const int*)d_in[0];    // [SEQ_LEN]
  const int*   current  = (const int*)d_in[1];    // [STATE_LEN]
  const float* cid_time = (const float*)d_in[2];  // [N_CID, N_TIME]
  float*       out      = (float*)d_out;          // [STATE_LEN, SEQ_LEN]

  const int state_len = in_sizes[1];              // 4096 rows -> 4096 blocks
  const int n_time    = N_TIME_C;                 // fixed by reference

  attn_gather_softmax_kernel<<<state_len, BLOCK, 0, stream>>>(
      history, current, cid_time, out, n_time);
}